// RFtoGPU_70385924046879
// MI455X (gfx1250) — compile-verified
//
#include <hip/hip_runtime.h>
#include <hip/hip_bf16.h>
#include <stdint.h>

// ---------------------------------------------------------------------------
// Random-forest inference, MI455X (gfx1250).
//   trees=128, depth=12 (exactly 12 unconditional steps: internal nodes never
//   have feat==-2), features=256, batch=100000.
// Strategy:
//   * TDM #1: 256-sample x 256-feature f32 tile -> LDS with hardware-padded
//     257-float row stride (bank-spread).
//   * TDM #2: top 6 tree levels (64 nodes x 64 trees x 8B = 32 KB) -> LDS,
//     strided tile gather done by the DMA engine. Halves L2 gather traffic.
//   * Nodes packed to 8B {feat,thresh}, tree stride 4096 (shift-only math).
//   * 8 independent tree-walk chains per thread (MLP at ~2 waves/SIMD).
//   * Trees split across 2 blocks per sample tile; deterministic combine.
// ---------------------------------------------------------------------------

#define N_TREES     128
#define N_NODES     8191      // 2^13 - 1
#define N_INTERNAL  4095      // 2^12 - 1
#define N_FEAT      256
#define PK_STRIDE   4096      // padded tree stride in packed node array
#define ROWSTRIDE   257       // LDS row stride in floats (256 + 1 pad)
#define TILE        256       // samples per block
#define TREE_SPLIT  2         // blocks per sample tile
#define TPB         (N_TREES / TREE_SPLIT)   // trees per block = 64
#define CHAINS      8         // concurrent tree walks per thread
#define LDS_LEVELS  6         // tree levels cached in LDS (nodes 0..62)
#define TOPN        64        // per-tree top-node slots in LDS (pow2)

#define XS_BYTES    ((size_t)TILE * ROWSTRIDE * sizeof(float))   // 263168
#define LN_BYTES    ((size_t)TPB * TOPN * 8)                     // 32768
#define SHMEM_BYTES (XS_BYTES + LN_BYTES)                        // 295936 < 320K

struct __attribute__((aligned(8))) Node {
    int   f;   // feature index
    float t;   // threshold
};

typedef __attribute__((ext_vector_type(4))) unsigned int uint32x4;
typedef __attribute__((ext_vector_type(8))) int          int32x8;
typedef __attribute__((ext_vector_type(4))) int          int32x4;

#if defined(__has_builtin)
#if __has_builtin(__builtin_amdgcn_tensor_load_to_lds) && __has_builtin(__builtin_amdgcn_s_wait_tensorcnt)
#define HAVE_TDM 1
#endif
#endif
#ifndef HAVE_TDM
#define HAVE_TDM 0
#endif

#if HAVE_TDM
// Generic 2D TDM load: rows x dim0 elements (element size 1<<ds_code bytes),
// global row stride `stride` elements, into LDS at byte offset lds_off.
// If pad==true, insert +1 DWORD into LDS after every 256 DWORDs stored.
__device__ __forceinline__ void tdm_load_2d(const void* gptr, unsigned lds_off,
                                            int ds_code, int dim0, int rows,
                                            int stride, bool pad) {
    unsigned long long ga = (unsigned long long)(uintptr_t)gptr;
    uint32x4 g0;
    g0[0] = 1u;                                   // count=1, user descriptor
    g0[1] = lds_off;                              // lds_addr (bytes)
    g0[2] = (unsigned)(ga & 0xFFFFFFFFu);         // global_addr[31:0]
    g0[3] = (unsigned)((ga >> 32) & 0x1FFFFFFu)   // global_addr[56:32]
          | (2u << 30);                           // type = 2 ("image")
    int32x8 g1;
    g1[0] = (ds_code << 16)                       // data_size
          | (pad ? ((1 << 20) | (7 << 22)) : 0);  // pad_enable, 256 DW, +1 DW
    g1[1] = (dim0 & 0xFFFF) << 16;                // tensor_dim0 lo16
    g1[2] = (rows & 0xFFFF) << 16;                // tensor_dim0 hi=0 | tensor_dim1 lo
    g1[3] = (dim0 & 0xFFFF) << 16;                // tensor_dim1 hi=0 | tile_dim0
    g1[4] = (rows & 0xFFFF);                      // tile_dim1 = rows, tile_dim2 = 0
    g1[5] = stride;                               // tensor_dim0_stride lo32
    g1[6] = 0;
    g1[7] = 0;
    int32x4 z4 = {0, 0, 0, 0};
#if defined(__clang_major__) && (__clang_major__ >= 23)
    int32x8 z8 = {0, 0, 0, 0, 0, 0, 0, 0};
    __builtin_amdgcn_tensor_load_to_lds(g0, g1, z4, z4, z8, 0);
#else
    __builtin_amdgcn_tensor_load_to_lds(g0, g1, z4, z4, 0);
#endif
}
#endif // HAVE_TDM

// ---------------------------------------------------------------------------
// Kernel 1: pack {feat, thresh} for internal nodes into 8B records.
// ---------------------------------------------------------------------------
__global__ __launch_bounds__(256)
void rf_pack_kernel(const int* __restrict__ feat, const float* __restrict__ thresh,
                    Node* __restrict__ pk) {
    int i = blockIdx.x * 256 + threadIdx.x;       // [0, 128*4096)
    int t = i >> 12;
    int n = i & (PK_STRIDE - 1);
    if (n < N_INTERNAL) {
        size_t src = (size_t)t * N_NODES + n;
        Node nd;
        nd.f = feat[src];
        nd.t = thresh[src];
        pk[i] = nd;
    }
}

// ---------------------------------------------------------------------------
// Stage X tile (always) into xs; TDM path used when available.
// ---------------------------------------------------------------------------
__device__ __forceinline__ void stage_x_fallback(float* xs, const float* __restrict__ X,
                                                 int s0, int rows) {
    for (int k = threadIdx.x; k < rows * N_FEAT; k += 256) {
        int r = k >> 8, c = k & (N_FEAT - 1);
        xs[r * ROWSTRIDE + c] = X[(size_t)s0 * N_FEAT + k];
    }
}

// ---------------------------------------------------------------------------
// Kernel 2: traversal. blockIdx.x -> sample tile, blockIdx.y -> tree half.
// ---------------------------------------------------------------------------
__global__ __launch_bounds__(256)
void rf_traverse_kernel(const float* __restrict__ X, const Node* __restrict__ pk,
                        const float* __restrict__ val, float* __restrict__ partial,
                        int batch) {
    extern __shared__ char smem[];
    float* xs     = (float*)smem;
    Node*  lnodes = (Node*)(smem + XS_BYTES);     // [TPB][TOPN]

    const int s0    = blockIdx.x * TILE;
    const int rows  = min(TILE, batch - s0);
    const int tree0 = blockIdx.y * TPB;
    const Node* pkb = pk + ((size_t)tree0 << 12);

#if HAVE_TDM
    if (threadIdx.x < 32) {                       // one wave drives the DMA
        // X tile: rows x 256 f32, padded to 257-float LDS stride.
        tdm_load_2d(X + (size_t)s0 * N_FEAT, 0, 2, N_FEAT, rows, N_FEAT, true);
        // Top-level node tile: TPB rows of TOPN 8-byte nodes, stride 4096.
        tdm_load_2d(pkb, (unsigned)XS_BYTES, 3, TOPN, TPB, PK_STRIDE, false);
        __builtin_amdgcn_s_wait_tensorcnt(0);
    }
#else
    stage_x_fallback(xs, X, s0, rows);
    for (int k = threadIdx.x; k < TPB * TOPN; k += 256) {
        int tl = k >> 6, n = k & (TOPN - 1);
        lnodes[k] = pkb[((size_t)tl << 12) + n];
    }
#endif
    __syncthreads();

    const float* myrow = xs + threadIdx.x * ROWSTRIDE;
    float acc = 0.0f;

#pragma unroll 1
    for (int t = 0; t < TPB; t += CHAINS) {
        int n[CHAINS];
#pragma unroll
        for (int j = 0; j < CHAINS; ++j) n[j] = 0;

        // Levels 0..5: nodes 0..62 served from LDS.
#pragma unroll
        for (int d = 0; d < LDS_LEVELS; ++d) {
#pragma unroll
            for (int j = 0; j < CHAINS; ++j) {
                Node a = lnodes[((t + j) << 6) + n[j]];
                n[j] = 2 * n[j] + 1 + (myrow[a.f] > a.t);
            }
        }
        // Levels 6..11: deep nodes gathered from L2 (8 independent chains).
#pragma unroll
        for (int d = LDS_LEVELS; d < 12; ++d) {
#pragma unroll
            for (int j = 0; j < CHAINS; ++j) {
                Node a = pkb[((size_t)(t + j) << 12) + n[j]];
                n[j] = 2 * n[j] + 1 + (myrow[a.f] > a.t);
            }
        }
#pragma unroll
        for (int j = 0; j < CHAINS; ++j)
            acc += val[(size_t)(tree0 + t + j) * N_NODES + n[j]];
    }

    const int s = s0 + threadIdx.x;
    if (s < batch)
        partial[(size_t)blockIdx.y * batch + s] = acc;
}

// ---------------------------------------------------------------------------
// Kernel 3: deterministic combine (no FP atomics).
// ---------------------------------------------------------------------------
__global__ __launch_bounds__(256)
void rf_combine_kernel(const float* __restrict__ partial, float* __restrict__ out,
                       int batch) {
    int i = blockIdx.x * 256 + threadIdx.x;
    if (i < batch) {
        float s = partial[i] + partial[(size_t)batch + i];
        out[i] = s * (1.0f / (float)N_TREES);
    }
}

// ---------------------------------------------------------------------------
// Fallback: single-pass traversal without workspace (all 128 trees per block,
// unpacked feat/thresh loads, direct write of the mean).
// ---------------------------------------------------------------------------
__global__ __launch_bounds__(256)
void rf_traverse_simple_kernel(const float* __restrict__ X,
                               const int* __restrict__ feat,
                               const float* __restrict__ thresh,
                               const float* __restrict__ val,
                               float* __restrict__ out, int batch) {
    extern __shared__ char smem[];
    float* xs = (float*)smem;
    const int s0   = blockIdx.x * TILE;
    const int rows = min(TILE, batch - s0);
#if HAVE_TDM
    if (threadIdx.x < 32) {
        tdm_load_2d(X + (size_t)s0 * N_FEAT, 0, 2, N_FEAT, rows, N_FEAT, true);
        __builtin_amdgcn_s_wait_tensorcnt(0);
    }
#else
    stage_x_fallback(xs, X, s0, rows);
#endif
    __syncthreads();

    const float* myrow = xs + threadIdx.x * ROWSTRIDE;
    float acc = 0.0f;
#pragma unroll 1
    for (int t = 0; t < N_TREES; t += 4) {
        size_t b[4];
        int n[4];
#pragma unroll
        for (int j = 0; j < 4; ++j) { b[j] = (size_t)(t + j) * N_NODES; n[j] = 0; }
#pragma unroll
        for (int d = 0; d < 12; ++d) {
#pragma unroll
            for (int j = 0; j < 4; ++j) {
                int   f0 = feat[b[j] + n[j]];
                float t0 = thresh[b[j] + n[j]];
                n[j] = 2 * n[j] + 1 + (myrow[f0] > t0);
            }
        }
#pragma unroll
        for (int j = 0; j < 4; ++j) acc += val[b[j] + n[j]];
    }
    const int s = s0 + threadIdx.x;
    if (s < batch)
        out[s] = acc * (1.0f / (float)N_TREES);
}

// ---------------------------------------------------------------------------
extern "C" void kernel_launch(void* const* d_in, const int* in_sizes, int n_in,
                              void* d_out, int out_size, void* d_ws, size_t ws_size,
                              hipStream_t stream) {
    const float* X      = (const float*)d_in[0];
    const int*   feat   = (const int*)d_in[1];
    const float* thresh = (const float*)d_in[2];
    // d_in[3] (left) / d_in[4] (right) are implied by 2n+1 / 2n+2 -> unused.
    const float* value  = (const float*)d_in[5];
    float*       out    = (float*)d_out;

    const int batch = in_sizes[0] / N_FEAT;                 // 100000
    const int nb    = (batch + TILE - 1) / TILE;            // 391

    const size_t pack_bytes    = (size_t)N_TREES * PK_STRIDE * sizeof(Node); // 4 MiB
    const size_t partial_bytes = (size_t)TREE_SPLIT * batch * sizeof(float);

    if (ws_size >= pack_bytes + partial_bytes) {
        Node*  pk      = (Node*)d_ws;
        float* partial = (float*)((char*)d_ws + pack_bytes);

        rf_pack_kernel<<<(N_TREES * PK_STRIDE) / 256, 256, 0, stream>>>(feat, thresh, pk);

        dim3 grid(nb, TREE_SPLIT);
        rf_traverse_kernel<<<grid, 256, SHMEM_BYTES, stream>>>(X, pk, value, partial,
                                                               batch);

        rf_combine_kernel<<<nb, 256, 0, stream>>>(partial, out, batch);
    } else {
        rf_traverse_simple_kernel<<<nb, 256, XS_BYTES, stream>>>(X, feat, thresh,
                                                                 value, out, batch);
    }
}